// EncoderMemNN_65747359367945
// MI455X (gfx1250) — compile-verified
//
#include <hip/hip_runtime.h>
#include <hip/hip_bf16.h>

#define B 32
#define M 512
#define T 16
#define D 128
#define V 32000

typedef __attribute__((ext_vector_type(2))) float v2f;
typedef __attribute__((ext_vector_type(8))) float v8f;

// ---------------------------------------------------------------------------
// Kernel 1: positions[b,m] = cumsum over m of (story[m,b,0] != PAD), pad -> 0.
// Also zeros u. One block per batch, 512 threads (one per memory slot).
// ---------------------------------------------------------------------------
__global__ void pos_kernel(const int* __restrict__ story,
                           int* __restrict__ positions,
                           float* __restrict__ u)
{
    int b = blockIdx.x;
    int m = threadIdx.x;                       // 0..511
    __shared__ int scan[M];
    int nonpad = (story[(size_t)m * (B * T) + (size_t)b * T] != 0) ? 1 : 0;
    scan[m] = nonpad;
    __syncthreads();
    // Hillis-Steele inclusive scan over M
    for (int off = 1; off < M; off <<= 1) {
        int v = scan[m];
        int w = (m >= off) ? scan[m - off] : 0;
        __syncthreads();
        scan[m] = v + w;
        __syncthreads();
    }
    positions[b * M + m] = nonpad ? scan[m] : 0;
    if (m < D) u[b * D + m] = 0.0f;
}

// ---------------------------------------------------------------------------
// Kernel 2: S_k[b,m,d] = sum_t C_k[story[m,b,t]][d] + pos_emb[positions[b,m]][d]
// for k = 0..3.  One wave32 per (b,m) slot, 8 slots per 256-thread block.
// Each lane owns 4 consecutive d's -> every embedding-row read is a single
// global_load_b128 per table (32 lanes x 16B = full 512B row). Token ids are
// broadcast inside the wave with __shfl (no LDS, no block sync).
// ---------------------------------------------------------------------------
__global__ void pool_kernel(const int* __restrict__ story,
                            const float* __restrict__ C_tables,
                            const float* __restrict__ pos_emb,
                            const int* __restrict__ positions,
                            float* __restrict__ SA)      // [4][B][M][D]
{
    int wave = threadIdx.x >> 5;
    int lane = threadIdx.x & 31;
    int slot = blockIdx.x * 8 + wave;   // b*M + m, 0..B*M-1
    int b = slot >> 9;                  // / M
    int m = slot & (M - 1);             // % M

    // lane < 16 loads one token id; broadcast later via shfl
    int tokval = 0;
    if (lane < T) tokval = story[(size_t)m * (B * T) + (size_t)b * T + lane];

    const float4* t0 = (const float4*)(C_tables);
    const float4* t1 = (const float4*)(C_tables + (size_t)1 * V * D);
    const float4* t2 = (const float4*)(C_tables + (size_t)2 * V * D);
    const float4* t3 = (const float4*)(C_tables + (size_t)3 * V * D);
    const int D4 = D / 4;               // 32 float4 per row; lane = d4 index

    float4 a0 = {0.f, 0.f, 0.f, 0.f};
    float4 a1 = a0, a2 = a0, a3 = a0;
#pragma unroll
    for (int t = 0; t < T; ++t) {
        int tok = __shfl(tokval, t, 32);
        size_t off = (size_t)tok * D4 + lane;
        float4 v0 = t0[off];
        float4 v1 = t1[off];
        float4 v2 = t2[off];
        float4 v3 = t3[off];
        a0.x += v0.x; a0.y += v0.y; a0.z += v0.z; a0.w += v0.w;
        a1.x += v1.x; a1.y += v1.y; a1.z += v1.z; a1.w += v1.w;
        a2.x += v2.x; a2.y += v2.y; a2.z += v2.z; a2.w += v2.w;
        a3.x += v3.x; a3.y += v3.y; a3.z += v3.z; a3.w += v3.w;
    }

    float4 pe = ((const float4*)pos_emb)[(size_t)positions[slot] * D4 + lane];
    a0.x += pe.x; a0.y += pe.y; a0.z += pe.z; a0.w += pe.w;
    a1.x += pe.x; a1.y += pe.y; a1.z += pe.z; a1.w += pe.w;
    a2.x += pe.x; a2.y += pe.y; a2.z += pe.z; a2.w += pe.w;
    a3.x += pe.x; a3.y += pe.y; a3.z += pe.z; a3.w += pe.w;

    float4* o = (float4*)SA;
    size_t base = (size_t)slot * D4 + lane;
    const size_t S4 = (size_t)B * M * D4;
    o[base + 0 * S4] = a0;
    o[base + 1 * S4] = a1;
    o[base + 2 * S4] = a2;
    o[base + 3 * S4] = a3;
}

// ---------------------------------------------------------------------------
// Kernel 3: one attention hop for one batch per block (256 thr = 8 wave32).
//   logits[m] = dot(S_A[b,m,:], u[b,:])   via V_WMMA_F32_16X16X4_F32
//   prob = softmax(logits)                block-wide reduction
//   o[d] = sum_m prob[m]*S_C[b,m,d]       via V_WMMA_F32_16X16X4_F32
//   u[b,d] += o[d]  (and -> out on final hop)
// WMMA broadcast-B trick: B[k][n] = vec[k] for all n; every output column is
// identical, so lanes 0/16 (N=0) carry rows 0-7 / 8-15 of the result.
// ---------------------------------------------------------------------------
__global__ void hop_kernel(const float* __restrict__ SA_A,
                           const float* __restrict__ SA_C,
                           float* __restrict__ u,
                           float* __restrict__ out)
{
    int b    = blockIdx.x;
    int tid  = threadIdx.x;           // 0..255
    int wave = tid >> 5;
    int lane = tid & 31;
    int row  = lane & 15;             // WMMA M index for A, column for D
    int khalf = lane >> 4;            // 0: K=0,1   1: K=2,3

    __shared__ float sh_u[D];
    __shared__ float sh_l[M];
    __shared__ float red[8];

    if (tid < D) sh_u[tid] = u[b * D + tid];
    __syncthreads();

    // ---- logits: 32 m-tiles of 16, 8 waves -> 4 tiles each; K-loop over D/4
    const float* Abase = SA_A + (size_t)b * M * D;
    for (int mt = wave * 4; mt < wave * 4 + 4; ++mt) {
        int m0 = mt * 16;
        v8f acc = {0.f, 0.f, 0.f, 0.f, 0.f, 0.f, 0.f, 0.f};
        for (int kc = 0; kc < D / 4; ++kc) {
            int d0 = kc * 4 + 2 * khalf;
            const float* p = Abase + (size_t)(m0 + row) * D + d0;
            v2f a;  a.x = p[0];        a.y = p[1];
            v2f bb; bb.x = sh_u[d0];   bb.y = sh_u[d0 + 1];
            acc = __builtin_amdgcn_wmma_f32_16x16x4_f32(
                false, a, false, bb, (short)0, acc, false, false);
        }
        if (row == 0) {
#pragma unroll
            for (int j = 0; j < 8; ++j)
                sh_l[m0 + khalf * 8 + j] = acc[j];
        }
    }
    __syncthreads();

    // ---- softmax over sh_l[0..511]
    float lmax = -__builtin_inff();
    for (int i = tid; i < M; i += 256) lmax = fmaxf(lmax, sh_l[i]);
    for (int off = 16; off > 0; off >>= 1)
        lmax = fmaxf(lmax, __shfl_xor(lmax, off, 32));
    if (lane == 0) red[wave] = lmax;
    __syncthreads();
    if (tid == 0) {
        float g = red[0];
        for (int i = 1; i < 8; ++i) g = fmaxf(g, red[i]);
        red[0] = g;
    }
    __syncthreads();
    float gmax = red[0];

    float lsum = 0.f;
    for (int i = tid; i < M; i += 256) {
        float e = __expf(sh_l[i] - gmax);
        sh_l[i] = e;
        lsum += e;
    }
    for (int off = 16; off > 0; off >>= 1)
        lsum += __shfl_xor(lsum, off, 32);
    __syncthreads();                      // protect red[] reuse
    if (lane == 0) red[wave] = lsum;
    __syncthreads();
    if (tid == 0) {
        float s = 0.f;
        for (int i = 0; i < 8; ++i) s += red[i];
        red[0] = s;
    }
    __syncthreads();
    float inv = 1.0f / red[0];
    for (int i = tid; i < M; i += 256) sh_l[i] *= inv;
    __syncthreads();

    // ---- o[d] = sum_m prob[m] * S_C[b,m,d]; 8 d-tiles, one per wave; K = m/4
    const float* Cbase = SA_C + (size_t)b * M * D;
    int d_tile = wave * 16;
    {
        v8f acc = {0.f, 0.f, 0.f, 0.f, 0.f, 0.f, 0.f, 0.f};
        for (int kc = 0; kc < M / 4; ++kc) {
            int m0 = kc * 4 + 2 * khalf;
            const float* p = Cbase + (size_t)m0 * D + d_tile + row;
            v2f a;  a.x = p[0];        a.y = p[D];     // rows m0, m0+1
            v2f bb; bb.x = sh_l[m0];   bb.y = sh_l[m0 + 1];
            acc = __builtin_amdgcn_wmma_f32_16x16x4_f32(
                false, a, false, bb, (short)0, acc, false, false);
        }
        if (row == 0) {
#pragma unroll
            for (int j = 0; j < 8; ++j) {
                int d = d_tile + khalf * 8 + j;
                float nu = sh_u[d] + acc[j];
                u[b * D + d] = nu;
                if (out) out[b * D + d] = nu;
            }
        }
    }
}

// ---------------------------------------------------------------------------
extern "C" void kernel_launch(void* const* d_in, const int* in_sizes, int n_in,
                              void* d_out, int out_size, void* d_ws, size_t ws_size,
                              hipStream_t stream)
{
    const int*   story    = (const int*)d_in[0];     // (M,B,T) int32
    const float* C_tables = (const float*)d_in[1];   // (4,V,D) f32
    const float* pos_emb  = (const float*)d_in[2];   // (513,D) f32
    float*       out      = (float*)d_out;           // (B,D) f32

    char* ws = (char*)d_ws;
    int*   positions = (int*)ws;                                   // B*M ints
    float* u         = (float*)(ws + (size_t)B * M * sizeof(int)); // B*D f32
    size_t sa_off    = (((size_t)B * M * sizeof(int) +
                         (size_t)B * D * sizeof(float)) + 255) & ~(size_t)255;
    float* SA        = (float*)(ws + sa_off);                      // 4*B*M*D f32
    const size_t S   = (size_t)B * M * D;

    pos_kernel <<<B,          M,   0, stream>>>(story, positions, u);
    pool_kernel<<<B * M / 8,  256, 0, stream>>>(story, C_tables, pos_emb,
                                                positions, SA);
    hop_kernel <<<B,          256, 0, stream>>>(SA + 0 * S, SA + 1 * S, u, nullptr);
    hop_kernel <<<B,          256, 0, stream>>>(SA + 1 * S, SA + 2 * S, u, nullptr);
    hop_kernel <<<B,          256, 0, stream>>>(SA + 2 * S, SA + 3 * S, u, out);
}